// MultiBoxLoss_24111946400557
// MI455X (gfx1250) — compile-verified
//
#include <hip/hip_runtime.h>
#include <math.h>

#define B_ 32
#define P_ 24564
#define C_ 81
#define O_ 8
#define THRESH 0.5f
#define NEG_RATIO 3

typedef float v2f __attribute__((ext_vector_type(2)));
typedef float v8f __attribute__((ext_vector_type(8)));
typedef unsigned int u32x4 __attribute__((ext_vector_type(4)));
typedef int i32x4 __attribute__((ext_vector_type(4)));
typedef int i32x8 __attribute__((ext_vector_type(8)));

// ---------------- K0: zero accumulators ----------------
__global__ void k_init(unsigned long long* packed, int* n_pos, float* fscal) {
    int i = blockIdx.x * blockDim.x + threadIdx.x;
    if (i < B_ * O_) packed[i] = 0ull;
    if (i < B_)      n_pos[i]  = 0;
    if (i < 4)       fscal[i]  = 0.0f;   // [0]=loc_sum [1]=conf_pos [2]=hard_neg [3]=n_pos_total(int bits)
}

// ---------------- K1: IoU matching ----------------
__global__ void k_match(const float* __restrict__ boxes, const float* __restrict__ priors,
                        float* __restrict__ ov_best, int* __restrict__ obj_best,
                        unsigned long long* __restrict__ packed) {
    int p = blockIdx.x * blockDim.x + threadIdx.x;
    int b = blockIdx.y;
    if (p >= P_) return;
    float4 pr = ((const float4*)priors)[p];          // cx,cy,w,h
    float px0 = pr.x - pr.z * 0.5f, py0 = pr.y - pr.w * 0.5f;
    float px1 = pr.x + pr.z * 0.5f, py1 = pr.y + pr.w * 0.5f;
    float parea = (px1 - px0) * (py1 - py0);
    float best = -1.0f; int bo = 0;
#pragma unroll
    for (int o = 0; o < O_; ++o) {
        float4 bx = ((const float4*)boxes)[b * O_ + o];  // x0,y0,x1,y1
        float ix0 = fmaxf(bx.x, px0), iy0 = fmaxf(bx.y, py0);
        float ix1 = fminf(bx.z, px1), iy1 = fminf(bx.w, py1);
        float inter = fmaxf(ix1 - ix0, 0.f) * fmaxf(iy1 - iy0, 0.f);
        float barea = (bx.z - bx.x) * (bx.w - bx.y);
        float iou = inter / (barea + parea - inter);
        if (iou > best) { best = iou; bo = o; }
        unsigned long long pk =
            (((unsigned long long)__float_as_uint(iou)) << 32) |
            (unsigned long long)(0xFFFFFFFFu - (unsigned)p);   // bigger key == smaller p on ties
        atomicMax(&packed[b * O_ + o], pk);
    }
    ov_best[b * P_ + p]  = best;
    obj_best[b * P_ + p] = bo;
}

// ---------------- K2: per-object override, sequential in o (last write wins) ----------------
__global__ void k_override(const unsigned long long* __restrict__ packed,
                           float* __restrict__ ov_best, int* __restrict__ obj_best) {
    int b = threadIdx.x;
    if (b >= B_) return;
    for (int o = 0; o < O_; ++o) {
        unsigned long long pk = packed[b * O_ + o];
        int p = (int)(0xFFFFFFFFu - (unsigned)(pk & 0xFFFFFFFFull));
        obj_best[b * P_ + p] = o;
        ov_best[b * P_ + p]  = 1.0f;
    }
}

// ---------------- K3a: labels, positive counts, localization L1 (WMMA wave reduction) --------
__global__ void k_loc(const float* __restrict__ pred_loc, const float* __restrict__ boxes,
                      const float* __restrict__ priors, const int* __restrict__ labels,
                      const float* __restrict__ ov_best, const int* __restrict__ obj_best,
                      int* __restrict__ lab_pp, int* __restrict__ n_pos,
                      int* __restrict__ n_pos_tot, float* __restrict__ loc_sum) {
    int p = blockIdx.x * blockDim.x + threadIdx.x;
    int b = blockIdx.y;
    bool valid = (p < P_);
    float part = 0.0f;
    bool pos = false;
    if (valid) {
        int idx = b * P_ + p;
        float ov = ov_best[idx];
        int o = obj_best[idx];
        int lab = (ov < THRESH) ? 0 : labels[b * O_ + o];
        lab_pp[idx] = lab;
        pos = (lab != 0);
        if (pos) {
            float4 bx = ((const float4*)boxes)[b * O_ + o];
            float4 pr = ((const float4*)priors)[p];
            float cx = 0.5f * (bx.x + bx.z), cy = 0.5f * (bx.y + bx.w);
            float w = bx.z - bx.x, h = bx.w - bx.y;
            float g0 = (cx - pr.x) / (pr.z * 0.1f);
            float g1 = (cy - pr.y) / (pr.w * 0.1f);
            float g2 = logf(w / pr.z) * 5.0f;
            float g3 = logf(h / pr.w) * 5.0f;
            float4 pl = ((const float4*)pred_loc)[idx];
            part = fabsf(pl.x - g0) + fabsf(pl.y - g1) + fabsf(pl.z - g2) + fabsf(pl.w - g3);
        }
    }
    int lane = threadIdx.x & 31;
    unsigned long long m = __ballot(pos);
    if (lane == 0) {
        int cnt = __popcll(m);
        if (cnt) { atomicAdd(&n_pos[b], cnt); atomicAdd(n_pos_tot, cnt); }
    }
    // Wave-wide sum of `part` through the matrix unit: A = 16x4 lane partials, B = ones.
    // D[m][n] = part[m] + part[m+16]; lane 0 holds rows 0..7, lane 16 rows 8..15.
    v2f a;  a.x = part;  a.y = 0.0f;
    v2f bm; bm.x = 1.0f; bm.y = 1.0f;
    v8f c = {0.f, 0.f, 0.f, 0.f, 0.f, 0.f, 0.f, 0.f};
    v8f d = __builtin_amdgcn_wmma_f32_16x16x4_f32(false, a, false, bm, (short)0, c, false, false);
    float s = d[0] + d[1] + d[2] + d[3] + d[4] + d[5] + d[6] + d[7];
    if ((lane == 0 || lane == 16) && s != 0.0f) atomicAdd(loc_sum, s);
}

// ---------------- K3b: one wave per prior: log-softmax CE ----------------
__global__ void k_ce(const float* __restrict__ scores, const int* __restrict__ lab_pp,
                     float* __restrict__ ce_neg, float* __restrict__ conf_pos) {
    long long wave = ((long long)blockIdx.x * blockDim.x + threadIdx.x) >> 5;
    int lane = threadIdx.x & 31;
    if (wave >= (long long)B_ * P_) return;
    const float* row = scores + (size_t)wave * C_;
    __builtin_prefetch(row + C_, 0, 0);                   // global_prefetch_b8
    float s0 = row[lane];                                 // lane       < 81 always
    float s1 = row[lane + 32];                            // lane + 32  < 81 always
    float s2 = (lane + 64 < C_) ? row[lane + 64] : -__builtin_inff();
    float mx = fmaxf(s0, fmaxf(s1, s2));
#pragma unroll
    for (int off = 16; off; off >>= 1) mx = fmaxf(mx, __shfl_xor(mx, off, 32));
    float e = expf(s0 - mx) + expf(s1 - mx) + ((lane + 64 < C_) ? expf(s2 - mx) : 0.f);
#pragma unroll
    for (int off = 16; off; off >>= 1) e += __shfl_xor(e, off, 32);
    if (lane == 0) {
        int lab = lab_pp[wave];
        float ce = logf(e) + mx - row[lab];
        if (lab != 0) { atomicAdd(conf_pos, ce); ce_neg[wave] = 0.0f; }
        else          { ce_neg[wave] = ce; }
    }
}

// ---------------- TDM: DMA one contiguous row of P_ floats from global into LDS ----------------
// D# built per CDNA5 ISA 8.3-8.5: 2-D tensor, tile = P_ x 1 elements of 4 bytes.
__device__ __forceinline__ void tdm_load_row(const float* gsrc, unsigned lds_off) {
#if __has_builtin(__builtin_amdgcn_tensor_load_to_lds) && __has_builtin(__builtin_amdgcn_s_wait_tensorcnt)
    unsigned long long ga = (unsigned long long)(size_t)gsrc;
    u32x4 g0;
    g0[0] = 1u;                                             // count=1 (valid), user mode
    g0[1] = lds_off;                                        // lds_addr[31:0]
    g0[2] = (unsigned)(ga & 0xFFFFFFFFull);                 // global_addr[31:0]
    g0[3] = (unsigned)((ga >> 32) & 0x01FFFFFFull)          // global_addr[56:32]
            | (2u << 30);                                   // type=2 ("image")
    i32x8 g1;
    g1[0] = (int)(2u << 16);                                // workgroup_mask=0, data_size=2 (4B)
    g1[1] = (int)(((unsigned)P_ & 0xFFFFu) << 16);          // tensor_dim0[15:0] @bits63:48
    g1[2] = (int)((((unsigned)P_ >> 16) & 0xFFFFu)          // tensor_dim0[31:16]
            | (1u << 16));                                  // tensor_dim1[15:0] = 1
    g1[3] = (int)(((unsigned)P_ & 0xFFFFu) << 16);          // tensor_dim1[31:16]=0, tile_dim0=P_
    g1[4] = 1;                                              // tile_dim1=1, tile_dim2=0
    g1[5] = (int)P_;                                        // tensor_dim0_stride[31:0]
    g1[6] = 0;                                              // stride0[47:32]=0, stride1[15:0]=0
    g1[7] = 0;
    i32x4 z4 = {0, 0, 0, 0};                                // groups 2/3 unused (2-D tensor)
#if __clang_major__ >= 23
    i32x8 z8 = {0, 0, 0, 0, 0, 0, 0, 0};
    __builtin_amdgcn_tensor_load_to_lds(g0, g1, z4, z4, z8, 0);
#else
    __builtin_amdgcn_tensor_load_to_lds(g0, g1, z4, z4, 0);
#endif
    __builtin_amdgcn_s_wait_tensorcnt(0);
#else
    // fallback: manual strided copy (wave 0 only reaches here; cover full row)
    float* dst = (float*)(size_t)lds_off;
    for (int i = threadIdx.x; i < P_; i += 32) dst[i] = gsrc[i];
#endif
}

// ---------------- K4: per-image top-K sum via LDS-resident bit binary search ----------------
__global__ void k_hardneg(const float* __restrict__ ce_neg, const int* __restrict__ n_pos,
                          float* __restrict__ hard_sum) {
    extern __shared__ float sm[];                         // P_ floats (96KB, fits 320KB LDS)
    __shared__ int   s_cnt;
    __shared__ float s_sum;
    int b = blockIdx.x;
    const float* src = ce_neg + (size_t)b * P_;
    if (threadIdx.x < 32) {                               // wave 0 issues the TDM DMA + waits
        tdm_load_row(src, (unsigned)(size_t)sm);
    }
    int K = NEG_RATIO * n_pos[b];
    if (K > P_) K = P_;
    __syncthreads();                                      // LDS tile visible to all waves
    if (K <= 0) return;                                   // uniform across block
    // all values >= 0 -> float bits are order-isomorphic to uint.
    // find smallest t with count(v > t) < K  (== the K-th largest value)
    unsigned lo = 0u, hi = 0x7F800000u;
    while (lo < hi) {
        unsigned mid = (lo + hi) >> 1;
        float t = __uint_as_float(mid);
        if (threadIdx.x == 0) s_cnt = 0;
        __syncthreads();
        int c = 0;
        for (int i = threadIdx.x; i < P_; i += blockDim.x) c += (sm[i] > t);
#pragma unroll
        for (int off = 16; off; off >>= 1) c += __shfl_xor(c, off, 32);
        if ((threadIdx.x & 31) == 0) atomicAdd(&s_cnt, c);
        __syncthreads();
        if (s_cnt < K) hi = mid; else lo = mid + 1;
        __syncthreads();
    }
    float t = __uint_as_float(lo);
    if (threadIdx.x == 0) { s_cnt = 0; s_sum = 0.f; }
    __syncthreads();
    float s = 0.f; int c = 0;
    for (int i = threadIdx.x; i < P_; i += blockDim.x) {
        float v = sm[i];
        if (v > t) { s += v; ++c; }
    }
#pragma unroll
    for (int off = 16; off; off >>= 1) { s += __shfl_xor(s, off, 32); c += __shfl_xor(c, off, 32); }
    if ((threadIdx.x & 31) == 0) { atomicAdd(&s_sum, s); atomicAdd(&s_cnt, c); }
    __syncthreads();
    if (threadIdx.x == 0) {
        // exact tie handling: top-K sum = sum(v>t) + (K - cnt) * t
        atomicAdd(hard_sum, s_sum + (float)(K - s_cnt) * t);
    }
}

// ---------------- K5: finalize ----------------
__global__ void k_final(const float* __restrict__ fscal, const int* __restrict__ n_pos_tot,
                        float* __restrict__ out) {
    if (threadIdx.x == 0 && blockIdx.x == 0) {
        float N = (float)(*n_pos_tot);
        out[0] = (fscal[2] + fscal[1]) / N + fscal[0] / (N * 4.0f);
    }
}

extern "C" void kernel_launch(void* const* d_in, const int* in_sizes, int n_in,
                              void* d_out, int out_size, void* d_ws, size_t ws_size,
                              hipStream_t stream) {
    const float* pred_loc = (const float*)d_in[0];   // [B,P,4]
    const float* scores   = (const float*)d_in[1];   // [B,P,C]
    const float* boxes    = (const float*)d_in[2];   // [B,O,4] xy
    const float* priors   = (const float*)d_in[3];   // [P,4] cxcy
    const int*   labels   = (const int*)d_in[4];     // [B,O]
    float* out = (float*)d_out;

    const size_t BP4 = (size_t)B_ * P_ * 4;
    char* ws = (char*)d_ws;
    float*              ce_neg   = (float*)(ws + 0 * BP4);
    int*                lab_pp   = (int*)  (ws + 1 * BP4);
    int*                obj_best = (int*)  (ws + 2 * BP4);
    float*              ov_best  = (float*)(ws + 3 * BP4);
    unsigned long long* packed   = (unsigned long long*)(ws + 4 * BP4);
    int*                n_pos    = (int*)  (ws + 4 * BP4 + 4096);
    float*              fscal    = (float*)(ws + 4 * BP4 + 4096 + 256);
    int*                npos_tot = (int*)(fscal + 3);

    dim3 gridP((P_ + 255) / 256, B_);

    k_init<<<1, 256, 0, stream>>>(packed, n_pos, fscal);
    k_match<<<gridP, 256, 0, stream>>>(boxes, priors, ov_best, obj_best, packed);
    k_override<<<1, 32, 0, stream>>>(packed, ov_best, obj_best);
    k_loc<<<gridP, 256, 0, stream>>>(pred_loc, boxes, priors, labels, ov_best, obj_best,
                                     lab_pp, n_pos, npos_tot, &fscal[0]);
    long long total_threads = (long long)B_ * P_ * 32;
    k_ce<<<(unsigned)((total_threads + 255) / 256), 256, 0, stream>>>(scores, lab_pp, ce_neg,
                                                                     &fscal[1]);
    k_hardneg<<<B_, 1024, P_ * sizeof(float), stream>>>(ce_neg, n_pos, &fscal[2]);
    k_final<<<1, 1, 0, stream>>>(fscal, npos_tot, out);
}